// LIF_61263413510487
// MI455X (gfx1250) — compile-verified
//
#include <hip/hip_runtime.h>
#include <cstdint>
#include <cstddef>

// LIF recurrence: u_t = u + (ir_t - u)/10 ; o_t = (u_t >= 1.0) ; u_next = u_t
// Shapes: ir [T=100, B=32, N=8192] f32, u0 [B,N] f32, out o_seq [T,B,N] f32.
// Pure streaming problem (~210 MB @ 23.3 TB/s => ~9us floor). Strategy:
// b128-vectorized coalesced stream; primary path uses CDNA5 async
// global->LDS loads (ASYNCcnt, depth-4 pipeline per wave); non-temporal
// hints so the one-shot 210 MB stream doesn't churn L2.

typedef float v4f __attribute__((ext_vector_type(4)));
typedef int   v4i __attribute__((ext_vector_type(4)));

// Address-space-qualified pointee types for the async-LDS builtin
// (signature per compiler diagnostic: (v4i AS1*, v4i AS3*, imm, imm)).
typedef __attribute__((address_space(1))) v4i v4i_gbl;
typedef __attribute__((address_space(3))) v4i v4i_lds;

#define LIF_T     100
#define LIF_BN    (32 * 8192)      /* 262144 neurons */
#define LIF_BN4   (LIF_BN / 4)     /* 65536 float4 lanes */
#define LIF_BLOCK 256
#define LIF_DEPTH 4                /* async pipeline stages (power of 2) */

#if defined(__has_builtin)
#if __has_builtin(__builtin_amdgcn_global_load_async_to_lds_b128) && \
    __has_builtin(__builtin_amdgcn_s_wait_asynccnt)
#define LIF_HAVE_ASYNC 1
#endif
#endif
#ifndef LIF_HAVE_ASYNC
#define LIF_HAVE_ASYNC 0
#endif

__device__ __forceinline__ v4f lif_update(v4f u, v4f x) {
  // u = u + (x - u) * 0.1  (matches reference arithmetic form)
  u.x = fmaf(x.x - u.x, 0.1f, u.x);
  u.y = fmaf(x.y - u.y, 0.1f, u.y);
  u.z = fmaf(x.z - u.z, 0.1f, u.z);
  u.w = fmaf(x.w - u.w, 0.1f, u.w);
  return u;
}

__device__ __forceinline__ v4f lif_spike(v4f u) {
  v4f o;
  o.x = (u.x >= 1.0f) ? 1.0f : 0.0f;
  o.y = (u.y >= 1.0f) ? 1.0f : 0.0f;
  o.z = (u.z >= 1.0f) ? 1.0f : 0.0f;
  o.w = (u.w >= 1.0f) ? 1.0f : 0.0f;
  return o;
}

#if LIF_HAVE_ASYNC
__device__ __forceinline__ void lif_async_load_b128(const v4f* g, v4f* l) {
  // global (AS1) -> LDS (AS3) async DMA, tracked by ASYNCcnt.
  __builtin_amdgcn_global_load_async_to_lds_b128(
      (v4i_gbl*)(void*)g,
      (v4i_lds*)(void*)l,
      /*imm offset*/ 0, /*cpol*/ 0);
}
#endif

// ---------------------------------------------------------------------------
// Primary: depth-4 async global->LDS pipeline. Each lane stages its own 16B
// into a private LDS slot; each wave waits only on its own ASYNCcnt, so no
// workgroup barriers are needed.
// ---------------------------------------------------------------------------
__global__ __launch_bounds__(LIF_BLOCK) void lif_async_kernel(
    const float* __restrict__ ir, const float* __restrict__ u0,
    float* __restrict__ out) {
#if LIF_HAVE_ASYNC
  __shared__ v4f buf[LIF_DEPTH][LIF_BLOCK];  // 16 KB / block
  const int tid = threadIdx.x;
  const int gid = blockIdx.x * LIF_BLOCK + tid;

  const v4f* ir4  = (const v4f*)ir;
  v4f*       out4 = (v4f*)out;
  v4f u = ((const v4f*)u0)[gid];

  // Prologue: fill the pipeline.
#pragma unroll
  for (int k = 0; k < LIF_DEPTH; ++k) {
    lif_async_load_b128(ir4 + (size_t)k * LIF_BN4 + gid, &buf[k][tid]);
  }

  for (int t = 0; t < LIF_T; ++t) {
    if (t + LIF_DEPTH < LIF_T) {
      // Async loads complete in order: <=DEPTH-1 outstanding means the
      // oldest stage (timestep t) has landed in LDS.
      __builtin_amdgcn_s_wait_asynccnt(LIF_DEPTH - 1);
    } else {
      __builtin_amdgcn_s_wait_asynccnt(0);  // drain the tail
    }
    asm volatile("" ::: "memory");  // LDS was written behind the compiler's back

    v4f x = buf[t & (LIF_DEPTH - 1)][tid];
    u = lif_update(u, x);
    __builtin_nontemporal_store(lif_spike(u), out4 + (size_t)t * LIF_BN4 + gid);

    const int tn = t + LIF_DEPTH;
    if (tn < LIF_T) {
      lif_async_load_b128(ir4 + (size_t)tn * LIF_BN4 + gid,
                          &buf[tn & (LIF_DEPTH - 1)][tid]);
    }
  }
#else
  (void)ir; (void)u0; (void)out;
#endif
}

// ---------------------------------------------------------------------------
// Fallback (also kept compiled for histogram feedback): plain non-temporal
// b128 stream with global_prefetch ~8 timesteps (8 MB machine-wide) ahead.
// ---------------------------------------------------------------------------
__global__ __launch_bounds__(LIF_BLOCK) void lif_plain_kernel(
    const float* __restrict__ ir, const float* __restrict__ u0,
    float* __restrict__ out) {
  const int gid = blockIdx.x * LIF_BLOCK + threadIdx.x;

  const v4f* ir4  = (const v4f*)ir;
  v4f*       out4 = (v4f*)out;
  v4f u = ((const v4f*)u0)[gid];

#pragma unroll 4
  for (int t = 0; t < LIF_T; ++t) {
    if (t + 8 < LIF_T) {
      __builtin_prefetch(ir4 + (size_t)(t + 8) * LIF_BN4 + gid, 0, 0);
    }
    v4f x = __builtin_nontemporal_load(ir4 + (size_t)t * LIF_BN4 + gid);
    u = lif_update(u, x);
    __builtin_nontemporal_store(lif_spike(u), out4 + (size_t)t * LIF_BN4 + gid);
  }
}

extern "C" void kernel_launch(void* const* d_in, const int* in_sizes, int n_in,
                              void* d_out, int out_size, void* d_ws,
                              size_t ws_size, hipStream_t stream) {
  (void)in_sizes; (void)n_in; (void)out_size; (void)d_ws; (void)ws_size;
  const float* ir = (const float*)d_in[0];  // [T,B,N]
  const float* u0 = (const float*)d_in[1];  // [B,N]
  float* out = (float*)d_out;               // [T,B,N]

  dim3 grid(LIF_BN4 / LIF_BLOCK);  // 256 blocks
  dim3 block(LIF_BLOCK);           // 256 threads = 8 wave32
#if LIF_HAVE_ASYNC
  lif_async_kernel<<<grid, block, 0, stream>>>(ir, u0, out);
#else
  lif_plain_kernel<<<grid, block, 0, stream>>>(ir, u0, out);
#endif
}